// CSS2D_30202210025558
// MI455X (gfx1250) — compile-verified
//
#include <hip/hip_runtime.h>
#include <math.h>

// Problem constants (match reference)
#define BB   2
#define LL   1024     // H*W = 32*32
#define DM   192      // d_model
#define DIN  384      // d_inner
#define NST  16       // d_state
#define RRK  12       // dt_rank
#define KDIR 4        // scan directions

typedef __attribute__((ext_vector_type(2))) float v2f;
typedef __attribute__((ext_vector_type(8))) float v8f;

// Index bijection for the 4 cross-scan directions over the 32x32 grid.
// k=0: identity; k=1: spatial transpose; k=2: flip; k=3: flip then transpose.
// The same map serves gather (build xs/xsc) and scatter (direction merge).
__device__ __forceinline__ int lmap(int k, int l) {
  int t = (k & 2) ? (1023 - l) : l;
  if (k & 1) t = ((t & 31) << 5) | (t >> 5);
  return t;
}

// ---------------------------------------------------------------------------
// Generic fp32 WMMA GEMM: C[M,N] = A[M,Kc] * W[N,Kc]^T   (one wave = one 16x16 tile)
// store_bdl==1: store C[(row>>10)*N + col][row&1023]  i.e. [b, channel, l]
// ---------------------------------------------------------------------------
__global__ void gemm_nt_wmma(const float* __restrict__ A, const float* __restrict__ W,
                             float* __restrict__ C, int M, int N, int Kc, int store_bdl) {
  int wave = (int)((blockIdx.x * blockDim.x + threadIdx.x) >> 5);
  int lane = (int)(threadIdx.x & 31);
  int ntiles = N >> 4;
  int tm = (wave / ntiles) << 4;
  int tn = (wave % ntiles) << 4;
  if (tm >= M) return;
  int half = lane >> 4;          // lane half selects K pair {0,1} vs {2,3}
  int l15  = lane & 15;
  const float* arow = A + (size_t)(tm + l15) * Kc + (half << 1);
  const float* brow = W + (size_t)(tn + l15) * Kc + (half << 1);
  v8f acc = {};
  for (int kk = 0; kk < Kc; kk += 4) {
    v2f af; af.x = arow[kk]; af.y = arow[kk + 1];
    v2f bf; bf.x = brow[kk]; bf.y = brow[kk + 1];
    acc = __builtin_amdgcn_wmma_f32_16x16x4_f32(false, af, false, bf,
                                                (short)0, acc, false, false);
  }
  int col = tn + l15;
#pragma unroll
  for (int r = 0; r < 8; ++r) {
    int row = tm + r + half * 8;
    if (store_bdl)
      C[((size_t)(row >> 10) * N + col) * 1024 + (row & 1023)] = acc[r];
    else
      C[(size_t)row * N + col] = acc[r];
  }
}

// ---------------------------------------------------------------------------
// Depthwise 3x3 conv + SiLU.  Input xp is the low half of xz [m,768]; output
// xh laid out [b, d, l] (scan-friendly).
// ---------------------------------------------------------------------------
__global__ void conv_silu_kernel(const float* __restrict__ xz, const float* __restrict__ conv_w,
                                 const float* __restrict__ conv_b, float* __restrict__ xh) {
  int idx = blockIdx.x * blockDim.x + threadIdx.x;   // (b*DIN + d)*LL + l
  if (idx >= BB * DIN * LL) return;
  int l = idx & 1023;
  int d = (idx >> 10) % DIN;
  int b = idx / (DIN * LL);
  int h = l >> 5, w = l & 31;
  float acc = conv_b[d];
#pragma unroll
  for (int i = 0; i < 3; ++i) {
    int hh = h + i - 1;
    if (hh < 0 || hh > 31) continue;
#pragma unroll
    for (int j = 0; j < 3; ++j) {
      int ww = w + j - 1;
      if (ww < 0 || ww > 31) continue;
      acc += conv_w[d * 9 + i * 3 + j] *
             xz[(size_t)((b << 10) + (hh << 5) + ww) * (2 * DIN) + d];
    }
  }
  xh[idx] = acc / (1.0f + __expf(-acc));   // SiLU
}

// ---------------------------------------------------------------------------
// x_dbl[b,k] (48x1024, rows 0..43 valid) = x_proj_weight[k] (44x384) @ Xsc(384x1024)
// where Xsc[d][l] = xcT[b,d,lmap(k,l)] (cross-scan gather fused into B loads).
// ---------------------------------------------------------------------------
__global__ void xdbl_wmma(const float* __restrict__ xcT, const float* __restrict__ xpw,
                          float* __restrict__ xdbl) {
  int wave = (int)((blockIdx.x * blockDim.x + threadIdx.x) >> 5);
  int lane = (int)(threadIdx.x & 31);
  int bk  = wave / 192;          // 3 m-tiles * 64 n-tiles per (b,k)
  int rem = wave % 192;
  if (bk >= BB * KDIR) return;
  int b = bk >> 2, k = bk & 3;
  int tm = (rem >> 6) << 4;      // 0,16,32
  int tn = (rem & 63) << 4;
  int half = lane >> 4, l15 = lane & 15;
  const float* Ab = xpw + (size_t)k * 44 * DIN;
  const float* Bb = xcT + (size_t)b * DIN * LL;
  int arow = tm + l15;
  bool aok = arow < 44;          // pad 44 -> 48 with zero rows
  int bl = lmap(k, tn + l15);    // per-lane gather column, loop invariant
  v8f acc = {};
  for (int kk = 0; kk < DIN; kk += 4) {
    int ka = kk + (half << 1);
    v2f af;
    af.x = aok ? Ab[(size_t)arow * DIN + ka]     : 0.0f;
    af.y = aok ? Ab[(size_t)arow * DIN + ka + 1] : 0.0f;
    v2f bf;
    bf.x = Bb[(size_t)ka * LL + bl];
    bf.y = Bb[(size_t)(ka + 1) * LL + bl];
    acc = __builtin_amdgcn_wmma_f32_16x16x4_f32(false, af, false, bf,
                                                (short)0, acc, false, false);
  }
  float* out = xdbl + (size_t)bk * 48 * LL;
  int col = tn + l15;
#pragma unroll
  for (int r = 0; r < 8; ++r)
    out[(size_t)(tm + r + half * 8) * LL + col] = acc[r];
}

// ---------------------------------------------------------------------------
// delta[b,k] (384x1024) = softplus( dt_projs_weight[k] (384x12) @ x_dbl_rows[0:12] + bias )
// ---------------------------------------------------------------------------
__global__ void dt_softplus_wmma(const float* __restrict__ xdbl, const float* __restrict__ dtw,
                                 const float* __restrict__ dtb, float* __restrict__ delta) {
  int wave = (int)((blockIdx.x * blockDim.x + threadIdx.x) >> 5);
  int lane = (int)(threadIdx.x & 31);
  int bk  = wave / 1536;         // 24 m-tiles * 64 n-tiles per (b,k)
  int rem = wave % 1536;
  if (bk >= BB * KDIR) return;
  int k = bk & 3;
  int tm = (rem >> 6) << 4;
  int tn = (rem & 63) << 4;
  int half = lane >> 4, l15 = lane & 15;
  const float* Ab = dtw + ((size_t)k * DIN + (tm + l15)) * RRK + (half << 1);
  const float* Bb = xdbl + (size_t)bk * 48 * LL;
  int col = tn + l15;
  v8f acc = {};
#pragma unroll
  for (int kk = 0; kk < RRK; kk += 4) {   // K = 12 = 3 WMMA steps
    v2f af; af.x = Ab[kk]; af.y = Ab[kk + 1];
    int kb = kk + (half << 1);
    v2f bf; bf.x = Bb[(size_t)kb * LL + col]; bf.y = Bb[(size_t)(kb + 1) * LL + col];
    acc = __builtin_amdgcn_wmma_f32_16x16x4_f32(false, af, false, bf,
                                                (short)0, acc, false, false);
  }
  float* out = delta + (size_t)bk * DIN * LL;
#pragma unroll
  for (int r = 0; r < 8; ++r) {
    int row = tm + r + half * 8;
    float v = acc[r] + dtb[k * DIN + row];
    v = (v > 20.0f) ? v : log1pf(__expf(v));       // softplus
    out[(size_t)row * LL + col] = v;
  }
}

// ---------------------------------------------------------------------------
// Selective scan.  One 16-lane half-wave per (b,k,d) channel; lane = state n.
// h_l = exp(delta_l * A_n) * h_{l-1} + delta_l * u_l * B_{n,l};  y_l = sum_n C_{n,l} h_l
// Output written directly at the merged position p = lmap(k,l) (per-k plane),
// with the skip term Ds*u folded in.
// ---------------------------------------------------------------------------
__global__ void scan_kernel(const float* __restrict__ delta, const float* __restrict__ xh,
                            const float* __restrict__ xdbl, const float* __restrict__ A_logs,
                            const float* __restrict__ Ds, float* __restrict__ yp) {
  int t = blockIdx.x * blockDim.x + threadIdx.x;
  int ci = t >> 4;               // channel = (b,k,d)
  int n  = t & 15;               // state index
  if (ci >= BB * KDIR * DIN) return;
  int d = ci % DIN;
  int k = (ci / DIN) & 3;
  int b = ci / (KDIR * DIN);
  int bk = b * KDIR + k;
  float An = -__expf(A_logs[(size_t)(k * DIN + d) * NST + n]);
  float Dv = Ds[k * DIN + d];
  const float* dP = delta + ((size_t)bk * DIN + d) * LL;
  const float* uP = xh    + ((size_t)b  * DIN + d) * LL;
  const float* BP = xdbl  + ((size_t)bk * 48 + 12 + n) * LL;
  const float* CP = xdbl  + ((size_t)bk * 48 + 28 + n) * LL;
  float* yP = yp + ((size_t)bk * DIN + d) * LL;
  float h = 0.0f;
  for (int l = 0; l < LL; ++l) {
    if ((l & 63) == 0) {                       // gfx1250 global_prefetch_b8
      __builtin_prefetch(dP + l + 64, 0, 0);
      __builtin_prefetch(BP + l + 64, 0, 0);
      __builtin_prefetch(CP + l + 64, 0, 0);
    }
    int p = lmap(k, l);                        // gather u and scatter y at same p
    float dl = dP[l];
    float u  = uP[p];
    float Bv = BP[l];
    float Cv = CP[l];
    h = __expf(dl * An) * h + (dl * u) * Bv;
    float y = Cv * h;
    y += __shfl_xor(y, 1, 32);                 // reduce over 16 states
    y += __shfl_xor(y, 2, 32);
    y += __shfl_xor(y, 4, 32);
    y += __shfl_xor(y, 8, 32);
    if (n == 0) yP[p] = y + Dv * u;
  }
}

// ---------------------------------------------------------------------------
// Sum 4 direction planes + LayerNorm over 384 channels + SiLU(z) gate.
// One block per token m = b*1024 + p.
// ---------------------------------------------------------------------------
__global__ void merge_ln_gate(const float* __restrict__ yp, const float* __restrict__ xz,
                              const float* __restrict__ g, const float* __restrict__ bbias,
                              float* __restrict__ yg) {
  __shared__ float s1[128], s2[128];
  int m = blockIdx.x;
  int b = m >> 10;
  int p = m & 1023;
  int tid = threadIdx.x;         // 128 threads, 3 channels each
  float vals[3];
  float sum = 0.f, sumsq = 0.f;
#pragma unroll
  for (int i = 0; i < 3; ++i) {
    int d = tid + i * 128;
    float v = 0.f;
#pragma unroll
    for (int k = 0; k < KDIR; ++k)
      v += yp[((size_t)(b * KDIR + k) * DIN + d) * LL + p];
    vals[i] = v;
    sum += v; sumsq += v * v;
  }
  s1[tid] = sum; s2[tid] = sumsq;
  __syncthreads();
  for (int off = 64; off > 0; off >>= 1) {
    if (tid < off) { s1[tid] += s1[tid + off]; s2[tid] += s2[tid + off]; }
    __syncthreads();
  }
  float mu   = s1[0] * (1.0f / DIN);
  float var  = s2[0] * (1.0f / DIN) - mu * mu;
  float rstd = rsqrtf(var + 1e-5f);
#pragma unroll
  for (int i = 0; i < 3; ++i) {
    int d = tid + i * 128;
    float zv = xz[(size_t)m * (2 * DIN) + DIN + d];
    float sil = zv / (1.f + __expf(-zv));
    yg[(size_t)m * DIN + d] = ((vals[i] - mu) * rstd * g[d] + bbias[d]) * sil;
  }
}

// ---------------------------------------------------------------------------
extern "C" void kernel_launch(void* const* d_in, const int* in_sizes, int n_in,
                              void* d_out, int out_size, void* d_ws, size_t ws_size,
                              hipStream_t stream) {
  const float* x               = (const float*)d_in[0];
  const float* x_cross         = (const float*)d_in[1];
  const float* in_proj_w       = (const float*)d_in[2];
  const float* in_proj_cross_w = (const float*)d_in[3];
  const float* conv_w          = (const float*)d_in[4];
  const float* conv_b          = (const float*)d_in[5];
  const float* x_proj_weight   = (const float*)d_in[6];
  const float* dt_w            = (const float*)d_in[7];
  const float* dt_b            = (const float*)d_in[8];
  const float* A_logs          = (const float*)d_in[9];
  const float* Ds              = (const float*)d_in[10];
  const float* g               = (const float*)d_in[11];
  const float* bbias           = (const float*)d_in[12];
  const float* out_proj_w      = (const float*)d_in[13];
  float* out = (float*)d_out;

  float* ws    = (float*)d_ws;
  float* xz    = ws;                               // [2048, 768]
  float* xcT   = xz    + (size_t)2048 * 768;       // [b, d, l]
  float* xh    = xcT   + (size_t)BB * DIN * LL;    // [b, d, l]
  float* xdbl  = xh    + (size_t)BB * DIN * LL;    // [b, k, 48, 1024]
  float* delta = xdbl  + (size_t)BB * KDIR * 48 * LL;   // [b, k, 384, 1024]
  float* yp    = delta + (size_t)BB * KDIR * DIN * LL;  // [b, k, 384, 1024] (merged pos)
  float* yg    = yp    + (size_t)BB * KDIR * DIN * LL;  // [2048, 384]

  // 1) xz = x @ in_proj_w^T          M=2048 N=768 K=192
  { int tiles = (2048 / 16) * (768 / 16);
    gemm_nt_wmma<<<(tiles * 32 + 255) / 256, 256, 0, stream>>>(x, in_proj_w, xz, 2048, 768, 192, 0); }
  // 2) xcT = (x_cross @ in_proj_cross_w^T) stored [b,d,l]
  { int tiles = (2048 / 16) * (384 / 16);
    gemm_nt_wmma<<<(tiles * 32 + 255) / 256, 256, 0, stream>>>(x_cross, in_proj_cross_w, xcT, 2048, 384, 192, 1); }
  // 3) depthwise conv + SiLU -> xh [b,d,l]
  conv_silu_kernel<<<(BB * DIN * LL + 255) / 256, 256, 0, stream>>>(xz, conv_w, conv_b, xh);
  // 4) per-direction projection x_dbl (dt/B/C rows)
  xdbl_wmma<<<(BB * KDIR * 192 * 32 + 255) / 256, 256, 0, stream>>>(xcT, x_proj_weight, xdbl);
  // 5) dt projection + softplus -> delta
  dt_softplus_wmma<<<(BB * KDIR * 1536 * 32 + 255) / 256, 256, 0, stream>>>(xdbl, dt_w, dt_b, delta);
  // 6) selective scan (49152 lanes)
  scan_kernel<<<(BB * KDIR * DIN * 16 + 255) / 256, 256, 0, stream>>>(delta, xh, xdbl, A_logs, Ds, yp);
  // 7) merge directions + LayerNorm + SiLU gate
  merge_ln_gate<<<BB * LL, 128, 0, stream>>>(yp, xz, g, bbias, yg);
  // 8) out = yg @ out_proj_w^T       M=2048 N=192 K=384
  { int tiles = (2048 / 16) * (192 / 16);
    gemm_nt_wmma<<<(tiles * 32 + 255) / 256, 256, 0, stream>>>(yg, out_proj_w, out, 2048, 192, 384, 0); }
}